// TXCDRPos_90984587198479
// MI455X (gfx1250) — compile-verified
//
#include <hip/hip_runtime.h>
#include <hip/hip_bf16.h>

// Problem sizes (fixed by the reference)
#define B_     256
#define T_     64
#define DIN_   512
#define DSAE_  4096
#define K_     32

typedef float v2f __attribute__((ext_vector_type(2)));
typedef float v8f __attribute__((ext_vector_type(8)));

// ---------------------------------------------------------------------------
// Kernel 0: pesum[d] = sum_t pe[t,d]  (sinusoidal posemb collapsed over t)
// ---------------------------------------------------------------------------
__global__ __launch_bounds__(512) void pesum_kernel(float* __restrict__ pesum) {
    const int d = threadIdx.x;                 // 512 threads, one block
    // div[j] = exp(-(2j) * ln(10000)/512); dim d uses j = d>>1
    const float f = __expf(-(float)(d & ~1) * (9.210340371976184f / 512.0f));
    float s = 0.0f;
    #pragma unroll 8
    for (int t = 0; t < T_; ++t) {
        const float a = (float)t * f;
        s += (d & 1) ? __cosf(a) : __sinf(a);
    }
    pesum[d] = s;
}

// ---------------------------------------------------------------------------
// Kernel 1: xsum[b,d] = sum_t x[b,t,d] + pesum[d]
//   One float4 of d per thread -> global_load_b128 (full cacheline per wave).
// ---------------------------------------------------------------------------
__global__ __launch_bounds__(256) void xsum_kernel(const float* __restrict__ x,
                                                   const float* __restrict__ pesum,
                                                   float* __restrict__ xsum) {
    const int gid = blockIdx.x * 256 + threadIdx.x;   // B*DIN/4 = 32768 threads
    const int b   = gid >> 7;                         // 128 float4s per (b) row
    const int d4  = (gid & 127) * 4;
    const float* xp = x + (size_t)b * (T_ * DIN_) + d4;
    float4 s = make_float4(0.f, 0.f, 0.f, 0.f);
    #pragma unroll 8
    for (int t = 0; t < T_; ++t) {
        const float4 v = *(const float4*)(xp + (size_t)t * DIN_);
        s.x += v.x; s.y += v.y; s.z += v.z; s.w += v.w;
    }
    const float4 p = *(const float4*)(pesum + d4);
    s.x += p.x; s.y += p.y; s.z += p.z; s.w += p.w;
    *(float4*)(xsum + (size_t)b * DIN_ + d4) = s;
}

// ---------------------------------------------------------------------------
// Kernel 2: pre = xsum @ W_enc + b_enc  via V_WMMA_F32_16X16X4_F32 (exact f32)
//   block = 256 threads = 8 wave32; each wave -> one 16x16 output tile,
//   block covers 16 rows x 128 cols; grid = (256/16, 4096/128) = (16, 32).
//   A tile (16x512) staged in LDS with padded stride 516 (bank-conflict free).
// ---------------------------------------------------------------------------
__global__ __launch_bounds__(256) void enc_gemm_kernel(const float* __restrict__ xsum,
                                                       const float* __restrict__ Wenc,
                                                       const float* __restrict__ benc,
                                                       float* __restrict__ pre) {
    __shared__ float ldsA[16 * 516];
    const int tid    = threadIdx.x;
    const int m_base = blockIdx.x * 16;
    const int n_base = blockIdx.y * 128 + (tid >> 5) * 16;   // wave -> 16 cols
    const int lane   = tid & 31;

    // Stage A tile: rows [m_base, m_base+16), all K=512. Coalesced.
    for (int i = tid; i < 16 * DIN_; i += 256) {
        const int r = i >> 9, c = i & (DIN_ - 1);
        ldsA[r * 516 + c] = xsum[(size_t)(m_base + r) * DIN_ + c];
    }
    __syncthreads();

    const int lrow  = lane & 15;          // M (for A) / N (for B) within tile
    const int khalf = (lane >> 4) << 1;   // lanes 0-15 -> K+0/1, 16-31 -> K+2/3
    const float* Bcol = Wenc + n_base + lrow;

    v8f acc = {};
    for (int k = 0; k < DIN_; k += 4) {
        v2f a;                                            // A 16x4 fragment
        const float* ap = &ldsA[lrow * 516 + k + khalf];
        a.x = ap[0];
        a.y = ap[1];
        v2f b;                                            // B 4x16 fragment
        const int krow = k + khalf;
        b.x = Bcol[(size_t)krow * DSAE_];
        b.y = Bcol[(size_t)(krow + 1) * DSAE_];
        acc = __builtin_amdgcn_wmma_f32_16x16x4_f32(
            /*neg_a=*/false, a, /*neg_b=*/false, b,
            /*c_mod=*/(short)0, acc, /*reuse_a=*/false, /*reuse_b=*/false);
    }

    // D layout: VGPR v holds M=v (lanes 0-15) / M=v+8 (lanes 16-31), N=lane&15
    const int   n    = n_base + lrow;
    const float bias = benc[n];
    const int   mo   = (lane >> 4) * 8;
    #pragma unroll
    for (int v = 0; v < 8; ++v)
        pre[(size_t)(m_base + v + mo) * DSAE_ + n] = acc[v] + bias;
}

// ---------------------------------------------------------------------------
// Kernel 3: per-row top-32 (iterative argmax, lowest-index tie-break), then
//   rewrite the row in-place as z = scatter(relu(topk)). One block per b.
//   Each thread owns 16 contiguous candidates in registers.
// ---------------------------------------------------------------------------
__global__ __launch_bounds__(256) void topk_kernel(float* __restrict__ zbuf,
                                                   int*   __restrict__ tidx,
                                                   float* __restrict__ tval) {
    const int b    = blockIdx.x;
    const int tid  = threadIdx.x;
    const int base = tid * 16;
    float* row = zbuf + (size_t)b * DSAE_;

    float r[16];
    #pragma unroll
    for (int j = 0; j < 16; ++j) r[j] = row[base + j];

    __shared__ float sv[256];
    __shared__ int   si[256];
    __shared__ int   kIdx[K_];
    __shared__ float kVal[K_];

    const float NEG_INF = -__builtin_inff();

    for (int it = 0; it < K_; ++it) {
        float lv = NEG_INF; int li = 0;
        #pragma unroll
        for (int j = 0; j < 16; ++j)
            if (r[j] > lv) { lv = r[j]; li = base + j; }   // strict > keeps lowest idx
        sv[tid] = lv; si[tid] = li;
        __syncthreads();
        for (int s = 128; s > 0; s >>= 1) {
            if (tid < s) {
                const float ov = sv[tid + s]; const int oi = si[tid + s];
                if (ov > sv[tid] || (ov == sv[tid] && oi < si[tid])) {
                    sv[tid] = ov; si[tid] = oi;
                }
            }
            __syncthreads();
        }
        if (tid == 0) {
            const int wi = si[0];
            const float wv = sv[0];
            kIdx[it] = wi;
            kVal[it] = wv > 0.0f ? wv : 0.0f;              // relu
            tidx[b * K_ + it] = wi;
            tval[b * K_ + it] = wv > 0.0f ? wv : 0.0f;
        }
        __syncthreads();
        const int wi = kIdx[it];
        if ((wi >> 4) == tid) r[wi & 15] = NEG_INF;        // knock out winner
        __syncthreads();
    }

    // z = zeros, then scatter relu(v) at the 32 (distinct) winner indices
    #pragma unroll
    for (int j = 0; j < 16; ++j) row[base + j] = 0.0f;
    __syncthreads();
    if (tid < K_) row[kIdx[tid]] = kVal[tid];
}

// ---------------------------------------------------------------------------
// Kernel 4: sparse decode + per-batch loss partial. Dominates HBM traffic, so
//   all gathers are float4 (global_load_b128 = 512B/instruction/wave).
//   One block per b; flatten (t,d) into 8192 float4s -> 32 iters of 256 thr.
//   x_hat stores stay scalar (d_out+1 is only 4B aligned).
// ---------------------------------------------------------------------------
__global__ __launch_bounds__(256) void decode_kernel(const float* __restrict__ x,
                                                     const float* __restrict__ Wdec,
                                                     const float* __restrict__ bdec,
                                                     const int*   __restrict__ tidx,
                                                     const float* __restrict__ tval,
                                                     float* __restrict__ xhat,
                                                     float* __restrict__ lossp) {
    const int b   = blockIdx.x;
    const int tid = threadIdx.x;
    __shared__ int   sIdx[K_];
    __shared__ float sVal[K_];
    if (tid < K_) { sIdx[tid] = tidx[b * K_ + tid]; sVal[tid] = tval[b * K_ + tid]; }
    __syncthreads();

    const float* xb  = x    + (size_t)b * T_ * DIN_;
    float*       xhb = xhat + (size_t)b * T_ * DIN_;

    float lsum = 0.0f;
    for (int it = 0; it < (T_ * DIN_) / (4 * 256); ++it) {     // 32 iterations
        const int off = (it * 256 + tid) * 4;                  // float4-aligned
        float4 acc = *(const float4*)(bdec + off);
        #pragma unroll
        for (int k = 0; k < K_; ++k) {
            const float4 w = *(const float4*)(Wdec + (size_t)sIdx[k] * (T_ * DIN_) + off);
            const float v = sVal[k];
            acc.x += v * w.x; acc.y += v * w.y;
            acc.z += v * w.z; acc.w += v * w.w;
        }
        // hint the next iteration's row of the first gathered tile into cache
        __builtin_prefetch(Wdec + (size_t)sIdx[0] * (T_ * DIN_) + off + 1024, 0, 0);
        const float4 xv = *(const float4*)(xb + off);
        xhb[off]     = acc.x;
        xhb[off + 1] = acc.y;
        xhb[off + 2] = acc.z;
        xhb[off + 3] = acc.w;
        const float dx = acc.x - xv.x, dy = acc.y - xv.y;
        const float dz = acc.z - xv.z, dw = acc.w - xv.w;
        lsum += dx * dx + dy * dy + dz * dz + dw * dw;
    }

    __shared__ float red[256];
    red[tid] = lsum;
    __syncthreads();
    for (int s = 128; s > 0; s >>= 1) {
        if (tid < s) red[tid] += red[tid + s];
        __syncthreads();
    }
    if (tid == 0) lossp[b] = red[0];
}

// ---------------------------------------------------------------------------
// Kernel 5: deterministic final loss reduction: mean over (b,t) of per-batch
//   sums of squared error over d.
// ---------------------------------------------------------------------------
__global__ __launch_bounds__(256) void loss_final_kernel(const float* __restrict__ lossp,
                                                         float* __restrict__ out) {
    const int tid = threadIdx.x;
    __shared__ float red[256];
    red[tid] = lossp[tid];
    __syncthreads();
    for (int s = 128; s > 0; s >>= 1) {
        if (tid < s) red[tid] += red[tid + s];
        __syncthreads();
    }
    if (tid == 0) out[0] = red[0] * (1.0f / (float)(B_ * T_));
}

// ---------------------------------------------------------------------------
extern "C" void kernel_launch(void* const* d_in, const int* in_sizes, int n_in,
                              void* d_out, int out_size, void* d_ws, size_t ws_size,
                              hipStream_t stream) {
    const float* x    = (const float*)d_in[0];   // (B,T,DIN)
    const float* Wenc = (const float*)d_in[1];   // (DIN,DSAE)
    const float* Wdec = (const float*)d_in[2];   // (DSAE,T,DIN)
    const float* benc = (const float*)d_in[3];   // (DSAE)
    const float* bdec = (const float*)d_in[4];   // (T,DIN)

    float* out  = (float*)d_out;
    float* loss = out;                                         // [1]
    float* xhat = out + 1;                                     // [B*T*DIN]
    float* zbuf = out + 1 + (size_t)B_ * T_ * DIN_;            // [B*DSAE] (pre -> z in-place)

    // Workspace layout (floats): pesum | xsum | tidx(int) | tval | lossp
    float* ws     = (float*)d_ws;
    float* pesum  = ws;                                        // 512
    float* xsum   = ws + 512;                                  // B*DIN = 131072
    int*   tidx   = (int*)(ws + 512 + (size_t)B_ * DIN_);      // B*K ints
    float* tval   = ws + 512 + (size_t)B_ * DIN_ + B_ * K_;    // B*K
    float* lossp  = tval + (size_t)B_ * K_;                    // B

    pesum_kernel<<<1, 512, 0, stream>>>(pesum);
    xsum_kernel<<<(B_ * DIN_ / 4) / 256, 256, 0, stream>>>(x, pesum, xsum);
    dim3 g2(B_ / 16, DSAE_ / 128);
    enc_gemm_kernel<<<g2, 256, 0, stream>>>(xsum, Wenc, benc, zbuf);
    topk_kernel<<<B_, 256, 0, stream>>>(zbuf, tidx, tval);
    decode_kernel<<<B_, 256, 0, stream>>>(x, Wdec, bdec, tidx, tval, xhat, lossp);
    loss_final_kernel<<<1, 256, 0, stream>>>(lossp, loss);
}